// TCLineEstimator_10453950398501
// MI455X (gfx1250) — compile-verified
//
#include <hip/hip_runtime.h>
#include <math.h>

// ---------------- problem constants ----------------
#define NN   768
#define DD   128
#define H0   256
#define H1   362
#define H1P  384          // H1 padded to multiple of 64 (and 16)

typedef float v2f __attribute__((ext_vector_type(2)));
typedef float v8f __attribute__((ext_vector_type(8)));

// ---------------- packing kernels ----------------
__global__ void tcl_pack_cat(const float* __restrict__ s0,
                             const float* __restrict__ s1,
                             float* __restrict__ cat) {
  int idx = blockIdx.x * blockDim.x + threadIdx.x;      // over 768*256
  if (idx >= NN * (2 * DD)) return;
  int r = idx >> 8;            // /256
  int c = idx & 255;
  cat[idx] = (c < DD) ? s0[r * DD + c] : s1[r * DD + (c - DD)];
}

__global__ void tcl_pack_w1p(const float* __restrict__ W,   // [384, 362]
                             float* __restrict__ Wp,        // [384, 384]
                             const float* __restrict__ b1,  // [362]
                             float* __restrict__ b1p,       // [384]
                             const float* __restrict__ W2,  // [362]
                             float* __restrict__ W2p) {     // [384]
  int idx = blockIdx.x * blockDim.x + threadIdx.x;
  if (idx < H1P * H1P) {
    int r = idx / H1P;
    int c = idx - r * H1P;
    Wp[idx] = (c < H1) ? W[r * H1 + c] : 0.f;
  }
  if (idx < H1P) {
    b1p[idx] = (idx < H1) ? b1[idx] : 0.f;
    W2p[idx] = (idx < H1) ? W2[idx] : 0.f;
  }
}

// ---------------- WMMA fp32 GEMM, transposed store ----------------
// CT[H, NN] = (A[NN, K] @ B[K, H])^T ; one wave per 16(row)x64(col) tile.
// grid.x = NN/16 row tiles, grid.y = H/64 col groups, block = 32 threads.
// k unrolled by 2 (two 16x16x4 steps); all 8 B fragments + 2 A fragments are
// loaded into distinct registers before the WMMA burst so VMEM can overlap
// with the matrix pipe instead of s_wait_loadcnt 0 before every WMMA.
#if __has_builtin(__builtin_amdgcn_wmma_f32_16x16x4_f32)
#define WMMA4(a, b, c) __builtin_amdgcn_wmma_f32_16x16x4_f32(false, (a), false, (b), (short)0, (c), false, false)
#endif

__global__ __launch_bounds__(32)
void tcl_wmma_gemm_t(const float* __restrict__ A, int ldA, int K,
                     const float* __restrict__ B, int ldB,
                     float* __restrict__ CT) {
  const int rowBase = blockIdx.x * 16;
  const int colBase = blockIdx.y * 64;
  const int lane = threadIdx.x;
  const int m    = lane & 15;
  const int half = lane >> 4;

  v8f acc0 = {}; v8f acc1 = {}; v8f acc2 = {}; v8f acc3 = {};

  const float* __restrict__ arow = A + (size_t)(rowBase + m) * ldA;

#ifdef WMMA4
  for (int k = 0; k < K; k += 8) {
    const int k0 = k + 2 * half;          // ISA A/B layout: half-wave K split
    const int k1 = k0 + 4;

    v2f a0; a0.x = arow[k0]; a0.y = arow[k0 + 1];
    v2f a1; a1.x = arow[k1]; a1.y = arow[k1 + 1];

    const float* p00 = B + (size_t)k0 * ldB + colBase + m;
    const float* p01 = p00 + ldB;
    const float* p10 = B + (size_t)k1 * ldB + colBase + m;
    const float* p11 = p10 + ldB;

    v2f b00, b01, b02, b03, b10, b11, b12, b13;
    b00.x = p00[0];  b00.y = p01[0];
    b01.x = p00[16]; b01.y = p01[16];
    b02.x = p00[32]; b02.y = p01[32];
    b03.x = p00[48]; b03.y = p01[48];
    b10.x = p10[0];  b10.y = p11[0];
    b11.x = p10[16]; b11.y = p11[16];
    b12.x = p10[32]; b12.y = p11[32];
    b13.x = p10[48]; b13.y = p11[48];

    acc0 = WMMA4(a0, b00, acc0);
    acc1 = WMMA4(a0, b01, acc1);
    acc2 = WMMA4(a0, b02, acc2);
    acc3 = WMMA4(a0, b03, acc3);
    acc0 = WMMA4(a1, b10, acc0);
    acc1 = WMMA4(a1, b11, acc1);
    acc2 = WMMA4(a1, b12, acc2);
    acc3 = WMMA4(a1, b13, acc3);
  }
#else
  // scalar fallback (keeps compiling if builtin is absent)
  for (int k = 0; k < K; ++k) {
    #pragma unroll
    for (int r = 0; r < 8; ++r) {
      float av = A[(size_t)(rowBase + r + 8 * half) * ldA + k];
      const float* br = B + (size_t)k * ldB + colBase + m;
      acc0[r] = fmaf(av, br[0],  acc0[r]);
      acc1[r] = fmaf(av, br[16], acc1[r]);
      acc2[r] = fmaf(av, br[32], acc2[r]);
      acc3[r] = fmaf(av, br[48], acc3[r]);
    }
  }
#endif

  // C element layout: VGPR r holds M = r + 8*half, N = lane&15
  #pragma unroll
  for (int r = 0; r < 8; ++r) {
    int row = rowBase + r + 8 * half;
    CT[(size_t)(colBase + m +  0) * NN + row] = acc0[r];
    CT[(size_t)(colBase + m + 16) * NN + row] = acc1[r];
    CT[(size_t)(colBase + m + 32) * NN + row] = acc2[r];
    CT[(size_t)(colBase + m + 48) * NN + row] = acc3[r];
  }
}

// ---------------- pairwise relu-dot + logsumexp ----------------
// Block handles 4 consecutive j rows. hxT/hyT are [Hpad, NN] (transposed).
// t[j][i] = b2 + sum_h relu(hxT[h][i] + hyT[h][j] + b1[h]) * w2[h]
// lse[j]  = logsumexp_i t[j][i];  t0[j] = t[j][j]
__global__ __launch_bounds__(256)
void tcl_pair_lse(const float* __restrict__ hxT, const float* __restrict__ hyT,
                  const float* __restrict__ b1, const float* __restrict__ w2,
                  const float* __restrict__ b2, int Hpad,
                  float* __restrict__ t0_out, float* __restrict__ lse_out) {
  __shared__ float sc[4 * H1P];
  __shared__ float sw[H1P];
  __shared__ float red[256];

  const int j0  = blockIdx.x * 4;
  const int tid = threadIdx.x;

  for (int h = tid; h < Hpad; h += 256) {
    float bb = b1[h];
    sw[h] = w2[h];
    #pragma unroll
    for (int jj = 0; jj < 4; ++jj)
      sc[jj * H1P + h] = hyT[(size_t)h * NN + (j0 + jj)] + bb;
  }
  __syncthreads();

  const float bias = b2[0];
  float t[3][4];

  #pragma unroll
  for (int l = 0; l < 3; ++l) {
    const int i = tid + l * 256;
    const float* __restrict__ xp = hxT + i;
    float a0 = 0.f, a1 = 0.f, a2 = 0.f, a3 = 0.f;
    #pragma unroll 4
    for (int h = 0; h < Hpad; ++h) {
      float x = xp[(size_t)h * NN];     // lane-coalesced over i
      float w = sw[h];                  // LDS broadcast
      a0 = fmaf(fmaxf(x + sc[0 * H1P + h], 0.f), w, a0);
      a1 = fmaf(fmaxf(x + sc[1 * H1P + h], 0.f), w, a1);
      a2 = fmaf(fmaxf(x + sc[2 * H1P + h], 0.f), w, a2);
      a3 = fmaf(fmaxf(x + sc[3 * H1P + h], 0.f), w, a3);
    }
    t[l][0] = a0 + bias; t[l][1] = a1 + bias; t[l][2] = a2 + bias; t[l][3] = a3 + bias;
  }

  // per-j reductions: row max, then expsum -> logsumexp; grab diagonal
  for (int jj = 0; jj < 4; ++jj) {
    const int j = j0 + jj;
    if ((j & 255) == tid) t0_out[j] = t[j >> 8][jj];   // i == j holder

    float mloc = fmaxf(fmaxf(t[0][jj], t[1][jj]), t[2][jj]);
    red[tid] = mloc; __syncthreads();
    for (int s = 128; s > 0; s >>= 1) {
      if (tid < s) red[tid] = fmaxf(red[tid], red[tid + s]);
      __syncthreads();
    }
    const float mx = red[0]; __syncthreads();

    float es = expf(t[0][jj] - mx) + expf(t[1][jj] - mx) + expf(t[2][jj] - mx);
    red[tid] = es; __syncthreads();
    for (int s = 128; s > 0; s >>= 1) {
      if (tid < s) red[tid] += red[tid + s];
      __syncthreads();
    }
    if (tid == 0) lse_out[j] = mx + logf(red[0]);
    __syncthreads();
  }
}

// ---------------- final scalar combine ----------------
__global__ __launch_bounds__(256)
void tcl_finalize(const float* __restrict__ t0_0, const float* __restrict__ lse_0,
                  const float* __restrict__ t0_1, const float* __restrict__ lse_1,
                  float* __restrict__ out) {
  __shared__ float red[256];
  const int tid = threadIdx.x;
  float s = 0.f;
  for (int i = tid; i < NN; i += 256)
    s += (t0_0[i] - lse_0[i]) + (t0_1[i] - lse_1[i]);
  red[tid] = s; __syncthreads();
  for (int st = 128; st > 0; st >>= 1) {
    if (tid < st) red[tid] += red[tid + st];
    __syncthreads();
  }
  if (tid == 0) out[0] = red[0] / (float)NN + 2.f * logf((float)NN);
}

// ---------------- workspace layout (floats) ----------------
#define OFF_CAT   ((size_t)0)                       // 768*256
#define OFF_W1P   (OFF_CAT  + (size_t)NN * 2 * DD)  // 384*384
#define OFF_B1P   (OFF_W1P  + (size_t)H1P * H1P)
#define OFF_W2P   (OFF_B1P  + (size_t)H1P)
#define OFF_HX0   (OFF_W2P  + (size_t)H1P)          // [256,768] transposed
#define OFF_HY0   (OFF_HX0  + (size_t)H0 * NN)
#define OFF_HX1   (OFF_HY0  + (size_t)H0 * NN)      // [384,768] transposed
#define OFF_HY1   (OFF_HX1  + (size_t)H1P * NN)
#define OFF_T00   (OFF_HY1  + (size_t)H1P * NN)
#define OFF_LSE0  (OFF_T00  + (size_t)NN)
#define OFF_T01   (OFF_LSE0 + (size_t)NN)
#define OFF_LSE1  (OFF_T01  + (size_t)NN)

extern "C" void kernel_launch(void* const* d_in, const int* in_sizes, int n_in,
                              void* d_out, int out_size, void* d_ws, size_t ws_size,
                              hipStream_t stream) {
  (void)in_sizes; (void)n_in; (void)out_size; (void)ws_size;
  const float* s0   = (const float*)d_in[0];
  const float* s1   = (const float*)d_in[1];
  const float* s2   = (const float*)d_in[2];
  const float* W1_0 = (const float*)d_in[3];   // [256,256]
  const float* b1_0 = (const float*)d_in[4];   // [256]
  const float* W2_0 = (const float*)d_in[5];   // [256]
  const float* b2_0 = (const float*)d_in[6];   // [1]
  const float* W1_1 = (const float*)d_in[7];   // [384,362]
  const float* b1_1 = (const float*)d_in[8];   // [362]
  const float* W2_1 = (const float*)d_in[9];   // [362]
  const float* b2_1 = (const float*)d_in[10];  // [1]
  float* out = (float*)d_out;
  float* ws  = (float*)d_ws;

  // 1) pack concatenated input and zero-padded estimator-1 params
  tcl_pack_cat<<<(NN * 2 * DD + 255) / 256, 256, 0, stream>>>(s0, s1, ws + OFF_CAT);
  tcl_pack_w1p<<<(H1P * H1P + 255) / 256, 256, 0, stream>>>(
      W1_1, ws + OFF_W1P, b1_1, ws + OFF_B1P, W2_1, ws + OFF_W2P);

  // 2) first-stage GEMMs via WMMA (stored transposed [H, N])
  dim3 g0(NN / 16, H0 / 64), g1(NN / 16, H1P / 64);
  tcl_wmma_gemm_t<<<g0, 32, 0, stream>>>(s0, DD, DD, W1_0, H0, ws + OFF_HX0);
  tcl_wmma_gemm_t<<<g0, 32, 0, stream>>>(s1, DD, DD, W1_0 + (size_t)DD * H0, H0, ws + OFF_HY0);
  tcl_wmma_gemm_t<<<g1, 32, 0, stream>>>(ws + OFF_CAT, 2 * DD, 2 * DD,
                                         ws + OFF_W1P, H1P, ws + OFF_HX1);
  tcl_wmma_gemm_t<<<g1, 32, 0, stream>>>(s2, DD, DD,
                                         ws + OFF_W1P + (size_t)(2 * DD) * H1P, H1P,
                                         ws + OFF_HY1);

  // 3) pairwise relu-dot + per-row logsumexp (4 j's per block)
  tcl_pair_lse<<<NN / 4, 256, 0, stream>>>(ws + OFF_HX0, ws + OFF_HY0,
                                           b1_0, W2_0, b2_0, H0,
                                           ws + OFF_T00, ws + OFF_LSE0);
  tcl_pair_lse<<<NN / 4, 256, 0, stream>>>(ws + OFF_HX1, ws + OFF_HY1,
                                           ws + OFF_B1P, ws + OFF_W2P, b2_1, H1P,
                                           ws + OFF_T01, ws + OFF_LSE1);

  // 4) scalar combine: sum of both InfoNCE bounds
  tcl_finalize<<<1, 256, 0, stream>>>(ws + OFF_T00, ws + OFF_LSE0,
                                      ws + OFF_T01, ws + OFF_LSE1, out);
}